// LocalUTSNodeEncoder_8237747273754
// MI455X (gfx1250) — compile-verified
//
#include <hip/hip_runtime.h>
#include <math.h>

// LocalUTSNodeEncoder signature kernel for MI455X (gfx1250).
// One wave32 per neighborhood; Gram via V_WMMA_F32_16X16X4_F32.

#define KNN 5
#define M_NODES 30
#define MP 32            // padded neighborhood size (2x2 WMMA tiles)
#define DIM 128
#define UTS 8
#define BIGF 1.0e9f
#define ROW_STRIDE 130   // dwords; pad 128 -> 130 to avoid LDS bank conflicts
#define WAVES_PER_BLOCK 2

typedef float v2f __attribute__((ext_vector_type(2)));
typedef float v8f __attribute__((ext_vector_type(8)));

__global__ __launch_bounds__(WAVES_PER_BLOCK * 32)
void uts_sig_kernel(const float* __restrict__ H,
                    const int* __restrict__ hood_idx,
                    const int* __restrict__ hood_sizes,
                    float* __restrict__ out, int N) {
  __shared__ float tileS[WAVES_PER_BLOCK][MP * ROW_STRIDE];   // 32x128 f32 tile (padded)
  __shared__ float DmS[WAVES_PER_BLOCK][MP][MP + 1];          // masked distance matrix
  __shared__ float sqS[WAVES_PER_BLOCK][MP];                  // squared norms (Gram diag)

  const int wave = threadIdx.x >> 5;
  const int lane = threadIdx.x & 31;
  const int n = blockIdx.x * WAVES_PER_BLOCK + wave;
  const int n0 = (n < N) ? n : (N - 1);   // clamp; never early-return (uniform barriers)

  float* tile = tileS[wave];
  float (*Dw)[MP + 1] = DmS[wave];
  float* sqw = sqS[wave];

  // ---- gather neighborhood rows into LDS: one row per pass, 32 lanes x float4 ----
  const int* idxp = hood_idx + (long long)n0 * M_NODES;
  int myidx = (lane < M_NODES) ? idxp[lane] : 0;
  for (int r = 0; r < MP; ++r) {
    float4 v = make_float4(0.f, 0.f, 0.f, 0.f);
    if (r < M_NODES) {   // wave-uniform branch
      int idx = __shfl(myidx, r, 32);
      const float4* src = reinterpret_cast<const float4*>(H + (long long)idx * DIM);
      v = src[lane];     // 512B coalesced row read (L2-resident: H is 25.6MB < 192MB L2)
    }
    float* dst = tile + r * ROW_STRIDE + lane * 4;
    dst[0] = v.x; dst[1] = v.y; dst[2] = v.z; dst[3] = v.w;
  }
  __syncthreads();

  // ---- Gram = Hl * Hl^T via f32 WMMA: 2x2 tiles of 16x16, K swept 4 at a time ----
  // f32 16x16x4 A layout: lanes 0-15 hold (K=kk,kk+1), lanes 16-31 hold (K=kk+2,kk+3),
  // row index = lane&15.  B (4x16) layout mirrors it with col = lane&15, so for this
  // symmetric product each lane's A and B fragments are identical LDS reads.
  const int half = lane >> 4;
  const int lm = lane & 15;
  v8f c00 = {}, c01 = {}, c10 = {}, c11 = {};
  const float* rowA = tile + lm * ROW_STRIDE + 2 * half;          // rows 0..15
  const float* rowB = tile + (lm + 16) * ROW_STRIDE + 2 * half;   // rows 16..31
  #pragma unroll 4
  for (int kk = 0; kk < DIM; kk += 4) {
    v2f f0 = *reinterpret_cast<const v2f*>(rowA + kk);
    v2f f1 = *reinterpret_cast<const v2f*>(rowB + kk);
    c00 = __builtin_amdgcn_wmma_f32_16x16x4_f32(false, f0, false, f0, (short)0, c00, false, false);
    c01 = __builtin_amdgcn_wmma_f32_16x16x4_f32(false, f0, false, f1, (short)0, c01, false, false);
    c10 = __builtin_amdgcn_wmma_f32_16x16x4_f32(false, f1, false, f0, (short)0, c10, false, false);
    c11 = __builtin_amdgcn_wmma_f32_16x16x4_f32(false, f1, false, f1, (short)0, c11, false, false);
  }

  // ---- squared norms = Gram diagonal (tiles (0,0) and (1,1)) ----
  #pragma unroll
  for (int v = 0; v < 8; ++v) {
    int i0 = half * 8 + v;            // row of accumulator vgpr v for this lane-half
    if (lm == i0) { sqw[i0] = c00[v]; sqw[i0 + 16] = c11[v]; }
  }
  __syncthreads();

  // ---- distances + pairwise stats; masked D matrix to LDS ----
  const int size = hood_sizes[n0];
  float sumD = 0.f, sumD2 = 0.f, maxD = 0.f;
  auto do_tile = [&](const v8f& c, int ti, int tj) {
    #pragma unroll
    for (int v = 0; v < 8; ++v) {
      int i = ti * 16 + half * 8 + v;
      int j = tj * 16 + lm;
      float d2 = sqw[i] + sqw[j] - 2.0f * c[v];
      float d = sqrtf(fmaxf(d2, 0.f) + 1e-12f);
      bool valid = (i < size) && (j < size) && (i != j);
      Dw[i][j] = valid ? d : BIGF;
      sumD  += valid ? d : 0.f;
      sumD2 += valid ? d * d : 0.f;
      maxD = valid ? fmaxf(maxD, d) : maxD;
    }
  };
  do_tile(c00, 0, 0); do_tile(c01, 0, 1);
  do_tile(c10, 1, 0); do_tile(c11, 1, 1);
  __syncthreads();

  // ---- per-row 5 smallest distances (row = lane), branchless sorted insert ----
  float knn0 = BIGF, knn1 = BIGF, knn2 = BIGF, knn3 = BIGF, knn4 = BIGF;
  #pragma unroll 8
  for (int j = 0; j < MP; ++j) {
    float x = Dw[lane][j];
    float t4 = fminf(knn4, fmaxf(knn3, x));
    float t3 = fminf(knn3, fmaxf(knn2, x));
    float t2 = fminf(knn2, fmaxf(knn1, x));
    float t1 = fminf(knn1, fmaxf(knn0, x));
    float t0 = fminf(knn0, x);
    knn0 = t0; knn1 = t1; knn2 = t2; knn3 = t3; knn4 = t4;
  }

  // ---- wave reductions (knn column sums over valid rows + pair stats) ----
  float row_w = (lane < size) ? 1.f : 0.f;
  float s0 = knn0 * row_w, s1 = knn1 * row_w, s2 = knn2 * row_w,
        s3 = knn3 * row_w, s4 = knn4 * row_w;
  #pragma unroll
  for (int off = 16; off; off >>= 1) {
    s0 += __shfl_xor(s0, off, 32);
    s1 += __shfl_xor(s1, off, 32);
    s2 += __shfl_xor(s2, off, 32);
    s3 += __shfl_xor(s3, off, 32);
    s4 += __shfl_xor(s4, off, 32);
    sumD  += __shfl_xor(sumD, off, 32);
    sumD2 += __shfl_xor(sumD2, off, 32);
    maxD = fmaxf(maxD, __shfl_xor(maxD, off, 32));
  }

  float fcnt = (float)size;
  float inv_cnt = 1.f / fcnt;
  float inv_np = 1.f / (fcnt * (fcnt - 1.f));
  float dmean = sumD * inv_np;
  float dvar = sumD2 * inv_np - dmean * dmean;   // E[D^2]-E[D]^2 == E[(D-mean)^2]

  if (n < N && lane < UTS) {
    float val;
    if      (lane == 0) val = s0 * inv_cnt;
    else if (lane == 1) val = s1 * inv_cnt;
    else if (lane == 2) val = s2 * inv_cnt;
    else if (lane == 3) val = s3 * inv_cnt;
    else if (lane == 4) val = s4 * inv_cnt;
    else if (lane == 5) val = dmean;
    else if (lane == 6) val = dvar;
    else                val = maxD;
    if (size < 3) val = 0.f;
    out[(long long)n * UTS + lane] = val;
  }
}

extern "C" void kernel_launch(void* const* d_in, const int* in_sizes, int n_in,
                              void* d_out, int out_size, void* d_ws, size_t ws_size,
                              hipStream_t stream) {
  const float* H          = (const float*)d_in[0];
  const int*   hood_idx   = (const int*)d_in[1];
  const int*   hood_sizes = (const int*)d_in[2];
  float*       out        = (float*)d_out;
  const int N = in_sizes[2];   // number of nodes (hood_sizes count)
  const int blocks = (N + WAVES_PER_BLOCK - 1) / WAVES_PER_BLOCK;
  uts_sig_kernel<<<blocks, WAVES_PER_BLOCK * 32, 0, stream>>>(H, hood_idx, hood_sizes, out, N);
}